// GRUModel_12232066859379
// MI455X (gfx1250) — compile-verified
//
#include <hip/hip_runtime.h>
#include <hip/hip_bf16.h>
#include <math.h>

// ---------------------------------------------------------------------------
// GRU Newton-sweep solver for MI455X (gfx1250, wave32, WMMA).
//   T=512, IN=128, H=256, OUT=10, 5 Newton sweeps.
// v3: LDS-tiled WMMA GEMM (64x64 macro-tile, K-block 32, double-buffered)
//     with gfx1250 async global->LDS copies (GLOBAL_LOAD_ASYNC_TO_LDS_B128,
//     ASYNCcnt) -- builtin parameter types fixed per compiler diagnostic:
//     arg0 is an AS1 pointer to a 16-byte int vector.
// ---------------------------------------------------------------------------

#define T_STEPS 512
#define IN_DIM  128
#define HDIM    256
#define H3      768
#define OUT_DIM 10
#define NITER   5

// GEMM tiling
#define BM   64
#define BN   64
#define BK   32
#define LDP  36            // padded LDS row stride (floats): 16B-aligned rows,
                           // gcd(36,64)=4 -> 16 consecutive rows hit 16 banks

typedef float v2f __attribute__((ext_vector_type(2)));
typedef float v8f __attribute__((ext_vector_type(8)));

#if __has_builtin(__builtin_amdgcn_global_load_async_to_lds_b128) && \
    __has_builtin(__builtin_amdgcn_s_wait_asynccnt)
#define USE_ASYNC_LDS 1
#else
#define USE_ASYNC_LDS 0
#endif

__device__ __forceinline__ float sigm(float x) { return 1.0f / (1.0f + __expf(-x)); }

#if USE_ASYNC_LDS
typedef int v4i __attribute__((vector_size(16)));
typedef v4i __attribute__((address_space(1))) v4i_g;   // global: 64-bit ptr
typedef v4i __attribute__((address_space(3))) v4i_l;   // LDS: 32-bit ptr

__device__ __forceinline__ void async_cp_b128(const float* g, float* l) {
  v4i_g* gp = (v4i_g*)(uintptr_t)g;
  v4i_l* lp = (v4i_l*)(uint32_t)(uintptr_t)l;
  // (global src, lds dst, ioffset, cpol)
  __builtin_amdgcn_global_load_async_to_lds_b128(gp, lp, 0, 0);
}
#endif

// ---------------------------------------------------------------------------
// C[M][N] = A[M][K] @ W[N][K]^T (+ bias[N]) with V_WMMA_F32_16X16X4_F32.
// Block: 256 threads (8 waves) -> 64x64 C macro-tile; wave (wr=w&3, wc=w>>2)
// computes C sub-tiles (16*wr, 32*wc) and (16*wr, 32*wc+16), sharing its
// A-fragment across both. K-blocked by 32 through double-buffered LDS tiles.
// Fragment layouts per CDNA5 ISA 7.12.2:
//   A 16x4 : lane l -> row (l&15); vgpr v holds K = 2*(l>>4) + v
//   B 4x16 : lane l -> col (l&15); vgpr v holds K = 2*(l>>4) + v
//   C 16x16: vgpr v -> row v + 8*(l>>4), col (l&15)
// ---------------------------------------------------------------------------
__global__ void __launch_bounds__(256, 2)
wmma_gemm_nt(float* __restrict__ C, const float* __restrict__ A,
             const float* __restrict__ W, const float* __restrict__ bias,
             int M, int N, int K) {
  __shared__ __align__(16) float sA[2][BM * LDP];
  __shared__ __align__(16) float sW[2][BN * LDP];

  const int tid   = threadIdx.x;
  const int lane  = tid & 31;
  const int wave  = tid >> 5;
  const int nBN   = N / BN;
  const int bm    = (blockIdx.x / nBN) * BM;
  const int bn    = (blockIdx.x % nBN) * BN;
  const int mlane = lane & 15;
  const int half  = lane >> 4;
  const int wr    = wave & 3;        // M sub-tile
  const int wc    = wave >> 2;       // N sub-tile pair

  // Loader mapping: threads 0..127 stage A (64x32), 128..255 stage W (64x32);
  // each thread moves 4 float4 chunks (chunk = i*128+tt: row=chunk>>3, c4=chunk&7).
  const int tt    = tid & 127;
  const bool ldW  = tid >= 128;
  const float* gsrc = ldW ? (W + (size_t)bn * K) : (A + (size_t)bm * K);

  auto issue = [&](int buf, int k0) {
    float* dst = ldW ? sW[buf] : sA[buf];
#pragma unroll
    for (int i = 0; i < 4; ++i) {
      const int chunk = i * 128 + tt;
      const int row   = chunk >> 3;
      const int c4    = (chunk & 7) * 4;
      const float* g  = gsrc + (size_t)row * K + k0 + c4;
      float*       l  = dst + row * LDP + c4;
#if USE_ASYNC_LDS
      async_cp_b128(g, l);
#else
      *reinterpret_cast<float4*>(l) = *reinterpret_cast<const float4*>(g);
#endif
    }
  };

  v8f acc0 = {};
  v8f acc1 = {};
  const int nkb = K / BK;

  issue(0, 0);
  for (int kb = 0; kb < nkb; ++kb) {
    const int cur = kb & 1;
    if (kb + 1 < nkb) {
      issue(cur ^ 1, (kb + 1) * BK);
#if USE_ASYNC_LDS
      __builtin_amdgcn_s_wait_asynccnt(4);   // this thread's 4 loads for `cur` done
#endif
    } else {
#if USE_ASYNC_LDS
      __builtin_amdgcn_s_wait_asynccnt(0);
#endif
    }
    __syncthreads();                          // everyone's tile data visible

    const float* tA  = sA[cur] + (wr * 16 + mlane) * LDP + 2 * half;
    const float* tB0 = sW[cur] + (wc * 32 + mlane) * LDP + 2 * half;
    const float* tB1 = tB0 + 16 * LDP;
#pragma unroll
    for (int kk = 0; kk < BK; kk += 4) {
      v2f a, b0, b1;
      a.x  = tA[kk];  a.y  = tA[kk + 1];
      b0.x = tB0[kk]; b0.y = tB0[kk + 1];
      b1.x = tB1[kk]; b1.y = tB1[kk + 1];
      // (neg_a, A, neg_b, B, c_mod, C, reuse_a, reuse_b)
      acc0 = __builtin_amdgcn_wmma_f32_16x16x4_f32(false, a, false, b0,
                                                   (short)0, acc0, false, false);
      acc1 = __builtin_amdgcn_wmma_f32_16x16x4_f32(false, a, false, b1,
                                                   (short)0, acc1, false, false);
    }
    __syncthreads();                          // done reading before next overwrite
  }

  const int   c0  = bn + wc * 32 + mlane;
  const int   c1  = c0 + 16;
  const float bb0 = bias ? bias[c0] : 0.0f;
  const float bb1 = bias ? bias[c1] : 0.0f;
#pragma unroll
  for (int v = 0; v < 8; ++v) {
    const size_t row = (size_t)(bm + wr * 16 + half * 8 + v) * N;
    C[row + c0] = acc0[v] + bb0;
    C[row + c1] = acc1[v] + bb1;
  }
}

__global__ void zero_f32(float* __restrict__ p, int n) {
  int i = blockIdx.x * blockDim.x + threadIdx.x;
  if (i < n) p[i] = 0.0f;
}

// ---------------------------------------------------------------------------
// Per-timestep linearization coefficients (one block per tau, thread per j):
//   A_t = diag(d0) + diag(d1) Wz + diag(d2) Wn + diag(d3) Wr
//   b_t = f_t - A_t h_prev   (reuses hgates -> O(H) per step)
// ---------------------------------------------------------------------------
__global__ void __launch_bounds__(HDIM)
gru_coeff(const float* __restrict__ IG, const float* __restrict__ HG,
          const float* __restrict__ ST, const float* __restrict__ bias_n,
          float* __restrict__ B, float* __restrict__ D0, float* __restrict__ D1,
          float* __restrict__ D2, float* __restrict__ D3) {
  const int tau = blockIdx.x;
  const int j   = threadIdx.x;

  if (tau == 0) {                       // f(h0=0, x0): hgates are zero
    float r = sigm(IG[j]);
    float z = sigm(IG[HDIM + j]);
    float n = tanhf(IG[2 * HDIM + j] + r * bias_n[j]);
    B[j]  = (1.0f - z) * n;
    D0[j] = 0.0f; D1[j] = 0.0f; D2[j] = 0.0f; D3[j] = 0.0f;
    return;
  }

  const float* ig = IG + (size_t)tau * H3;
  const float* hg = HG + (size_t)(tau - 1) * H3;
  const float  hprev = ST[(size_t)(tau - 1) * HDIM + j];

  const float hgr = hg[j], hgz = hg[HDIM + j], hgn = hg[2 * HDIM + j];
  const float r = sigm(ig[j] + hgr);
  const float z = sigm(ig[HDIM + j] + hgz);
  const float m = hgn + bias_n[j];
  const float n = tanhf(ig[2 * HDIM + j] + r * m);
  const float f = n + z * (hprev - n);

  const float dz = z * (1.0f - z);
  const float dr = r * (1.0f - r);
  const float dn = 1.0f - n * n;

  const float d0 = z;
  const float d1 = (hprev - n) * dz;
  const float d2 = (1.0f - z) * dn * r;
  const float d3 = (1.0f - z) * dn * m * dr;

  const size_t o = (size_t)tau * HDIM + j;
  D0[o] = d0; D1[o] = d1; D2[o] = d2; D3[o] = d3;
  B[o]  = f - (d0 * hprev + d1 * hgz + d2 * hgn + d3 * hgr);
}

// ---------------------------------------------------------------------------
// Sequential affine scan: h_t = A_t h_{t-1} + b_t via g = W_hh h (row/thread).
// One persistent 768-thread workgroup; h double-buffered in LDS; next step's
// coefficient rows prefetched under the dot-product loop.
// ---------------------------------------------------------------------------
__global__ void __launch_bounds__(H3, 1)
gru_scan(const float* __restrict__ Whh, const float* __restrict__ B,
         const float* __restrict__ D0, const float* __restrict__ D1,
         const float* __restrict__ D2, const float* __restrict__ D3,
         float* __restrict__ ST) {
  __shared__ __align__(16) float hs[2][HDIM];
  __shared__ __align__(16) float g[H3];
  const int tid = threadIdx.x;

  if (tid < HDIM) {
    float v = B[tid];                   // new_states[0] = b_0 (A_0 = 0)
    hs[0][tid] = v;
    ST[tid] = v;
  }
  __syncthreads();

  const float4* wr = reinterpret_cast<const float4*>(Whh + (size_t)tid * HDIM);
  int p = 0;
  for (int tau = 1; tau < T_STEPS; ++tau) {
    if (tid < HDIM && tau + 1 < T_STEPS) {       // hide coeff-row latency
      const size_t o1 = (size_t)(tau + 1) * HDIM + tid;
      __builtin_prefetch(D0 + o1, 0, 0);
      __builtin_prefetch(D1 + o1, 0, 0);
      __builtin_prefetch(D2 + o1, 0, 0);
      __builtin_prefetch(D3 + o1, 0, 0);
      __builtin_prefetch(B  + o1, 0, 0);
    }

    const float4* hp = reinterpret_cast<const float4*>(hs[p]);
    float acc = 0.0f;
#pragma unroll 8
    for (int k = 0; k < HDIM / 4; ++k) {
      float4 w  = wr[k];
      float4 h4 = hp[k];
      acc += w.x * h4.x + w.y * h4.y + w.z * h4.z + w.w * h4.w;
    }
    g[tid] = acc;
    __syncthreads();

    if (tid < HDIM) {
      const size_t o = (size_t)tau * HDIM + tid;
      float hn = D0[o] * hs[p][tid] + D1[o] * g[HDIM + tid] +
                 D2[o] * g[2 * HDIM + tid] + D3[o] * g[tid] + B[o];
      // jnp.nan_to_num: NaN->0, +/-inf -> +/-FLT_MAX
      hn = (hn != hn) ? 0.0f : fminf(fmaxf(hn, -3.402823466e38f), 3.402823466e38f);
      hs[p ^ 1][tid] = hn;
      ST[o] = hn;
    }
    __syncthreads();
    p ^= 1;
  }
}

// out[o] = w_out[o] . h_last + b_out[o]; one wave32 per output row.
__global__ void __launch_bounds__(OUT_DIM * 32)
out_proj(const float* __restrict__ Wout, const float* __restrict__ bout,
         const float* __restrict__ ST, float* __restrict__ out) {
  const int lane = threadIdx.x & 31;
  const int w    = threadIdx.x >> 5;
  const float* h = ST + (size_t)(T_STEPS - 1) * HDIM;
  float acc = 0.0f;
  for (int j = lane; j < HDIM; j += 32) acc += Wout[w * HDIM + j] * h[j];
#pragma unroll
  for (int off = 16; off > 0; off >>= 1) acc += __shfl_down(acc, off, 32);
  if (lane == 0) out[w] = acc + bout[w];
}

extern "C" void kernel_launch(void* const* d_in, const int* in_sizes, int n_in,
                              void* d_out, int out_size, void* d_ws, size_t ws_size,
                              hipStream_t stream) {
  const float* X      = (const float*)d_in[0]; // (512,128)
  const float* Wih    = (const float*)d_in[1]; // (768,128)
  const float* Whh    = (const float*)d_in[2]; // (768,256)
  const float* bias   = (const float*)d_in[3]; // (768)
  const float* bias_n = (const float*)d_in[4]; // (256)
  const float* Wout   = (const float*)d_in[5]; // (10,256)
  const float* bout   = (const float*)d_in[6]; // (10)

  float* ws = (float*)d_ws;                    // ~6.3 MB of f32 scratch
  float* IG = ws;                              // 512*768
  float* HG = IG + (size_t)T_STEPS * H3;       // 512*768
  float* ST = HG + (size_t)T_STEPS * H3;       // 512*256 (states)
  float* Bv = ST + (size_t)T_STEPS * HDIM;     // 512*256
  float* D0 = Bv + (size_t)T_STEPS * HDIM;     // 4 x 512*256
  float* D1 = D0 + (size_t)T_STEPS * HDIM;
  float* D2 = D1 + (size_t)T_STEPS * HDIM;
  float* D3 = D2 + (size_t)T_STEPS * HDIM;

  const int gemmBlocks = (T_STEPS / BM) * (H3 / BN);   // 8 * 12 = 96

  // states0 = 0
  zero_f32<<<(T_STEPS * HDIM + 255) / 256, 256, 0, stream>>>(ST, T_STEPS * HDIM);

  // igates (input-only, once): IG = X @ Wih^T + bias   [M=512, N=768, K=128]
  wmma_gemm_nt<<<gemmBlocks, 256, 0, stream>>>(IG, X, Wih, bias,
                                               T_STEPS, H3, IN_DIM);

  for (int it = 0; it < NITER; ++it) {
    // hgates: HG = ST @ Whh^T   [M=512, N=768, K=256]
    wmma_gemm_nt<<<gemmBlocks, 256, 0, stream>>>(HG, ST, Whh, nullptr,
                                                 T_STEPS, H3, HDIM);
    gru_coeff<<<T_STEPS, HDIM, 0, stream>>>(IG, HG, ST, bias_n, Bv, D0, D1, D2, D3);
    gru_scan<<<1, H3, 0, stream>>>(Whh, Bv, D0, D1, D2, D3, ST);
  }

  out_proj<<<1, OUT_DIM * 32, 0, stream>>>(Wout, bout, ST, (float*)d_out);
}